// GCN_33200097198998
// MI455X (gfx1250) — compile-verified
//
#include <hip/hip_runtime.h>
#include <hip/hip_bf16.h>
#include <math.h>

// Problem constants (match reference)
#define NN 100000
#define NE 1600000
#define NG 512
#define F  100        // feature dim
#define H  200        // hidden dim
#define EPS_MSG 1e-7f
#define EPS_SM  1e-16f
#define BN_RSQRT 0.9999950000374997f   // 1/sqrt(1+1e-5)

typedef __attribute__((ext_vector_type(2))) float v2f;
typedef __attribute__((ext_vector_type(8))) float v8f;

// ---------------- elementwise helpers ----------------
__global__ void k_zero(float* __restrict__ p, int n) {
  int i = blockIdx.x * blockDim.x + threadIdx.x;
  if (i < n) p[i] = 0.f;
}

__global__ void k_count(const int* __restrict__ batch, float* __restrict__ cnt, int n) {
  int i = blockIdx.x * blockDim.x + threadIdx.x;
  if (i < n) atomicAdd(&cnt[batch[i]], 1.0f);
}

// zero-pad K=10 -> K=16 rows (for conv1 x / src_w / dst_w)
__global__ void k_pad16(const float* __restrict__ src, float* __restrict__ dst, int rows) {
  int i = blockIdx.x * blockDim.x + threadIdx.x;   // rows*16
  if (i >= rows * 16) return;
  int r = i >> 4, c = i & 15;
  dst[i] = (c < 10) ? src[r * 10 + c] : 0.f;
}

// num = num/(den+eps) + xd   (agg + residual)
__global__ void k_combine(float* __restrict__ num, const float* __restrict__ den,
                          const float* __restrict__ xd, int n) {
  int i = blockIdx.x * blockDim.x + threadIdx.x;
  if (i < n) num[i] = num[i] / (den[i] + EPS_SM) + xd[i];
}

// ---------------- WMMA fp32 GEMM ----------------
// C[M,Ncol] = A[M,K] * W[Ncol,K]^T (+bias, +BN, +ReLU).   Requires K%4==0, M%16==0.
// One wave computes a 16x64 C macro-tile: one A frag feeds 4 chained
// V_WMMA_F32_16X16X4_F32 per k-step (4 accumulators). All loads unconditional:
// B row pointers are clamped to Ncol-1 (garbage cols masked at store).
#define NT 4
__launch_bounds__(256)
__global__ void k_gemm_wmma(const float* __restrict__ A, int lda,
                            const float* __restrict__ W, int ldw,
                            float* __restrict__ C, int ldc,
                            int M, int Ncol, int K,
                            const float* __restrict__ bias,
                            const float* __restrict__ bng,
                            const float* __restrict__ bnb,
                            int relu) {
  const int tilesN = (Ncol + 63) >> 6;
  const int tilesM = M >> 4;
  int tile = blockIdx.x * 8 + (threadIdx.x >> 5);
  if (tile >= tilesM * tilesN) return;       // wave-uniform exit: EXEC stays full for WMMA
  const int lane = threadIdx.x & 31;
  const int tm = tile / tilesN;
  const int tn = tile - tm * tilesN;
  const int row0 = tm << 4, col0 = tn << 6;

  const int half = lane >> 4;                // 0: K=k0..k0+1, 1: K=k0+2..k0+3
  const int l15  = lane & 15;
  const int koff = half << 1;

  const float* Ap = A + (size_t)(row0 + l15) * lda + koff;
  const float* Wp[NT];
  int col[NT];
  #pragma unroll
  for (int t = 0; t < NT; ++t) {
    col[t] = col0 + t * 16 + l15;
    int cc = col[t] < Ncol ? col[t] : Ncol - 1;     // clamp: loads always in-bounds
    Wp[t] = W + (size_t)cc * ldw + koff;
  }

  v8f acc[NT];
  #pragma unroll
  for (int t = 0; t < NT; ++t) acc[t] = (v8f){0.f,0.f,0.f,0.f,0.f,0.f,0.f,0.f};

  #pragma unroll 2
  for (int k0 = 0; k0 < K; k0 += 4) {
    v2f a  = *(const v2f*)(Ap + k0);
    v2f b0 = *(const v2f*)(Wp[0] + k0);
    v2f b1 = *(const v2f*)(Wp[1] + k0);
    v2f b2 = *(const v2f*)(Wp[2] + k0);
    v2f b3 = *(const v2f*)(Wp[3] + k0);
    acc[0] = __builtin_amdgcn_wmma_f32_16x16x4_f32(false, a, false, b0, (short)0, acc[0], false, false);
    acc[1] = __builtin_amdgcn_wmma_f32_16x16x4_f32(false, a, false, b1, (short)0, acc[1], false, false);
    acc[2] = __builtin_amdgcn_wmma_f32_16x16x4_f32(false, a, false, b2, (short)0, acc[2], false, false);
    acc[3] = __builtin_amdgcn_wmma_f32_16x16x4_f32(false, a, false, b3, (short)0, acc[3], false, false);
  }

  #pragma unroll
  for (int t = 0; t < NT; ++t) {
    if (col[t] < Ncol) {
      float bi = bias ? bias[col[t]] : 0.f;
      float sc = bng ? (bng[col[t]] * BN_RSQRT) : 1.f;
      float sh = bng ? bnb[col[t]] : 0.f;
      #pragma unroll
      for (int j = 0; j < 8; ++j) {
        int row = row0 + j + (half << 3);   // C layout: VGPR j -> M=j (lanes 0-15), M=j+8 (16-31)
        float v = (acc[t][j] + bi) * sc + sh;
        if (relu) v = fmaxf(v, 0.f);
        C[(size_t)row * ldc + col[t]] = v;
      }
    }
  }
}

// ---------------- edge passes (softmax aggregation) ----------------
// 128 threads per edge (100 active features); 2 edges per 256-thread block.
__global__ void k_edge_max(const int* __restrict__ src, const int* __restrict__ dst,
                           const float* __restrict__ h, unsigned* __restrict__ mU) {
  int tid = threadIdx.x;
  int e = blockIdx.x * 2 + (tid >> 7);
  int f = tid & 127;
  if (e >= NE || f >= F) return;
  int s = src[e], d = dst[e];
  float msg = fmaxf(h[s * F + f], 0.f) + EPS_MSG;          // > 0 always
  atomicMax(&mU[d * F + f], __float_as_uint(msg));         // positive-float bit order
}

__global__ void k_edge_sum(const int* __restrict__ src, const int* __restrict__ dst,
                           const float* __restrict__ h, const float* __restrict__ m,
                           float* __restrict__ num, float* __restrict__ den) {
  int tid = threadIdx.x;
  int e = blockIdx.x * 2 + (tid >> 7);
  int f = tid & 127;
  if (e >= NE || f >= F) return;
  int s = src[e], d = dst[e];
  float msg = fmaxf(h[s * F + f], 0.f) + EPS_MSG;
  int idx = d * F + f;
  float ex = expf(msg - m[idx]);
  atomicAdd(&den[idx], ex);
  atomicAdd(&num[idx], ex * msg);
}

// ---------------- readout ----------------
__global__ void k_readout_acc(const float* __restrict__ h, const int* __restrict__ batch,
                              float* __restrict__ rsum, unsigned* __restrict__ rmaxU) {
  int tid = threadIdx.x;
  int node = blockIdx.x * 2 + (tid >> 7);
  int f = tid & 127;
  if (node >= NN || f >= F) return;
  int b = batch[node];
  float v = h[node * F + f];                               // v >= 0 (post-ReLU)
  atomicAdd(&rsum[b * F + f], v);
  atomicMax(&rmaxU[b * F + f], __float_as_uint(v));
}

__global__ void k_readout_fin(const float* __restrict__ rsum, const float* __restrict__ rmax,
                              const float* __restrict__ cnt, float* __restrict__ encode) {
  int i = blockIdx.x * blockDim.x + threadIdx.x;           // over NG*F
  if (i >= NG * F) return;
  int g = i / F, f = i - g * F;
  float c = fmaxf(cnt[g], 1.f);
  float mean = rsum[i] / c;
  encode[g * (2 * F) + f]     += fmaxf(mean, 0.f);         // relu(mean)
  encode[g * (2 * F) + F + f] += fmaxf(rmax[i], 0.f);      // relu(max)
}

// ---------------- classifier head (tiny: 512x2) ----------------
__global__ void k_cls(const float* __restrict__ z, const float* __restrict__ w,
                      const float* __restrict__ b, float* __restrict__ logits) {
  int t = blockIdx.x * blockDim.x + threadIdx.x;
  if (t >= NG * 2) return;
  int g = t >> 1, c = t & 1;
  float s = b[c];
  for (int k = 0; k < F; ++k) s += z[g * F + k] * w[c * F + k];
  logits[g * 2 + c] = s;
}

// ---------------- host orchestration ----------------
static inline int cdiv(int a, int b) { return (a + b - 1) / b; }

static void gemm(hipStream_t st, const float* A, int lda, const float* W, int ldw,
                 float* C, int ldc, int M, int Ncol, int K,
                 const float* bias, const float* g, const float* be, int relu) {
  int tiles = (M >> 4) * ((Ncol + 63) >> 6);
  k_gemm_wmma<<<cdiv(tiles, 8), 256, 0, st>>>(A, lda, W, ldw, C, ldc, M, Ncol, K, bias, g, be, relu);
}

extern "C" void kernel_launch(void* const* d_in, const int* in_sizes, int n_in,
                              void* d_out, int out_size, void* d_ws, size_t ws_size,
                              hipStream_t stream) {
  const float* x   = (const float*)d_in[0];
  const int*   ei  = (const int*)d_in[1];
  const int*   bat = (const int*)d_in[2];
  const int* src = ei;           // edge_index[0]
  const int* dst = ei + NE;      // edge_index[1]

  const float* c1_src_w = (const float*)d_in[3];  const float* c1_src_b = (const float*)d_in[4];
  const float* c1_dst_w = (const float*)d_in[5];  const float* c1_dst_b = (const float*)d_in[6];
  const float* cw1[3] = {(const float*)d_in[7],  (const float*)d_in[13], (const float*)d_in[19]};
  const float* cb1[3] = {(const float*)d_in[8],  (const float*)d_in[14], (const float*)d_in[20]};
  const float* cg [3] = {(const float*)d_in[9],  (const float*)d_in[15], (const float*)d_in[21]};
  const float* cbe[3] = {(const float*)d_in[10], (const float*)d_in[16], (const float*)d_in[22]};
  const float* cw2[3] = {(const float*)d_in[11], (const float*)d_in[17], (const float*)d_in[23]};
  const float* cb2[3] = {(const float*)d_in[12], (const float*)d_in[18], (const float*)d_in[24]};
  const float* fc1_w = (const float*)d_in[25]; const float* fc1_b = (const float*)d_in[26];
  const float* cls_w = (const float*)d_in[27]; const float* cls_b = (const float*)d_in[28];

  float* logits = (float*)d_out;            // [512,2]
  float* encode = (float*)d_out + NG * 2;   // [512,200] (accumulated here, reused as fc1 input)

  // Workspace layout (floats): 5*N*F + N*H + N*16 + small  ~ 288 MB
  const size_t NF = (size_t)NN * F;
  float* ws    = (float*)d_ws;
  float* feat0 = ws;                 // conv outputs (ping)
  float* feat1 = ws + NF;            // conv outputs (pong); conv1 xd aliased here
  float* m     = ws + 2 * NF;        // segment max
  float* num   = ws + 3 * NF;        // sum(e*msg) -> combined node features
  float* den   = ws + 4 * NF;        // sum(e)
  float* HH    = ws + 5 * NF;        // [N,200] MLP hidden; conv1 h_src aliased here
  float* rsum  = ws + 7 * NF;        // [G,100]
  float* rmax  = rsum + (size_t)NG * F;
  float* cnt   = rmax + (size_t)NG * F;
  float* zbuf  = cnt + NG;           // [G,100]
  float* xpad  = zbuf + (size_t)NG * F;          // [N,16]  x zero-padded K
  float* wsp   = xpad + (size_t)NN * 16;         // [100,16]
  float* wdp   = wsp + 100 * 16;                 // [100,16]

  const int T = 256;
  const int egrid = (NE + 1) / 2;
  const int ngrid = (NN + 1) / 2;

  // global init: encode=0, per-graph counts
  k_zero<<<cdiv(NG * 2 * F, T), T, 0, stream>>>(encode, NG * 2 * F);
  k_zero<<<cdiv(NG, T), T, 0, stream>>>(cnt, NG);
  k_count<<<cdiv(NN, T), T, 0, stream>>>(bat, cnt, NN);

  // conv1 K=10 -> pad to 16 so the GEMM inner loop needs no K masking
  k_pad16<<<cdiv(NN * 16, T), T, 0, stream>>>(x, xpad, NN);
  k_pad16<<<cdiv(100 * 16, T), T, 0, stream>>>(c1_src_w, wsp, 100);
  k_pad16<<<cdiv(100 * 16, T), T, 0, stream>>>(c1_dst_w, wdp, 100);

  for (int c = 0; c < 3; ++c) {
    const float *hsrc, *xd;
    float* outFeat;
    if (c == 0) {
      // conv1: h_src = x@src_w^T+b (into HH region), xd = x@dst_w^T+b (into feat1 region)
      gemm(stream, xpad, 16, wsp, 16, HH,    F, NN, F, 16, c1_src_b, nullptr, nullptr, 0);
      gemm(stream, xpad, 16, wdp, 16, feat1, F, NN, F, 16, c1_dst_b, nullptr, nullptr, 0);
      hsrc = HH; xd = feat1; outFeat = feat0;
    } else if (c == 1) {
      hsrc = feat0; xd = feat0; outFeat = feat1;
    } else {
      hsrc = feat1; xd = feat1; outFeat = feat0;
    }

    // softmax aggregation: m=max, then num/den sums, then combine with residual
    k_zero<<<cdiv((int)(3 * NF), T), T, 0, stream>>>(m, (int)(3 * NF));   // m,num,den contiguous
    k_edge_max<<<egrid, T, 0, stream>>>(src, dst, hsrc, (unsigned*)m);
    k_edge_sum<<<egrid, T, 0, stream>>>(src, dst, hsrc, m, num, den);
    k_combine<<<cdiv((int)NF, T), T, 0, stream>>>(num, den, xd, (int)NF);

    // MLP: Linear(100->200)+BN+ReLU, then Linear(200->100)+ReLU(outer relu fused)
    gemm(stream, num, F, cw1[c], F, HH,      H, NN, H, F, cb1[c], cg[c], cbe[c], 1);
    gemm(stream, HH,  H, cw2[c], H, outFeat, F, NN, F, H, cb2[c], nullptr, nullptr, 1);

    // readout: mean & max per graph, accumulate relu() into encode
    k_zero<<<cdiv(NG * 2 * F, T), T, 0, stream>>>(rsum, NG * 2 * F);      // rsum+rmax contiguous
    k_readout_acc<<<ngrid, T, 0, stream>>>(outFeat, bat, rsum, (unsigned*)rmax);
    k_readout_fin<<<cdiv(NG * F, T), T, 0, stream>>>(rsum, rmax, cnt, encode);
  }

  // head: z = relu(encode @ fc1^T + b); logits = z @ cls^T + b
  gemm(stream, encode, 2 * F, fc1_w, 2 * F, zbuf, F, NG, F, 2 * F, fc1_b, nullptr, nullptr, 1);
  k_cls<<<cdiv(NG * 2, T), T, 0, stream>>>(zbuf, cls_w, cls_b, logits);
}